// GeometrySerializer_27075473834734
// MI455X (gfx1250) — compile-verified
//
#include <hip/hip_runtime.h>
#include <stdint.h>

// ---------------- problem constants (fixed by the reference) ----------------
#define NB    4
#define NVIEW 6
#define NL    16384
#define NI    4096
#define NPTS  (NL + NI)        // 20480 points per batch
#define CH    256
#define MPAD  32768            // sort size (next pow2 of 20480)
#define LOGM  15
#define TILE  8192             // LDS sort tile (64KB of u64 keys)
#define KEYSHIFT 15
#define IDXMASK  32767u

// ---------------- workspace layout (bytes) ----------------
#define WS_KEYS   0                                 // NB*MPAD u64 = 1MB
#define WS_Q      (WS_KEYS + (size_t)NB*MPAD*8)     // NB*NPTS*3 i32
#define WS_MM     (WS_Q    + (size_t)NB*NPTS*3*4)   // NB*6 i32

// ---------------- output layout (floats, concatenated return order) --------
#define OUT_TOK   ((size_t)0)
#define OUT_IDX   ((size_t)NB*NPTS*CH)              // 20,971,520
#define OUT_MASK  (OUT_IDX  + (size_t)NB*NPTS)      // +81,920
#define OUT_SCAL  (OUT_MASK + (size_t)NB*NPTS)      // +81,920

// ============================================================ init
__global__ void k_init(int* __restrict__ mm, float* __restrict__ out) {
    int t = threadIdx.x;
    if (t < NB * 6) mm[t] = ((t % 6) < 3) ? INT_MAX : INT_MIN;
    if (t == NB * 6)     out[OUT_SCAL + 0] = (float)NL;
    if (t == NB * 6 + 1) out[OUT_SCAL + 1] = 1.0f;
}

// ============================================================ quantize
__device__ __forceinline__ void inv3(const float m[9], float o[9]) {
    float a=m[0],b=m[1],c=m[2],d=m[3],e=m[4],f=m[5],g=m[6],h=m[7],k=m[8];
    float A=(e*k-f*h), B=-(d*k-f*g), C=(d*h-e*g);
    float det = a*A + b*B + c*C;
    float r = 1.0f / det;
    o[0]=A*r;              o[1]=-(b*k-c*h)*r;   o[2]=(b*f-c*e)*r;
    o[3]=B*r;              o[4]=(a*k-c*g)*r;    o[5]=-(a*f-c*d)*r;
    o[6]=C*r;              o[7]=-(a*h-b*g)*r;   o[8]=(a*e-b*d)*r;
}

// Grid is exact and NPTS % 256 == 0, so each block is wholly inside one batch
// and all lanes are active -> safe wave32 shuffle reduction.
__global__ void __launch_bounds__(256)
k_quant(const float* __restrict__ lidar_coords, const float* __restrict__ img_kuvd,
        const float* __restrict__ Kc, const float* __restrict__ Tc,
        const float* __restrict__ Rp, const float* __restrict__ tp,
        int* __restrict__ q, int* __restrict__ mm) {
    int gid = blockIdx.x * 256 + threadIdx.x;
    int b = gid / NPTS, i = gid % NPTS;
    float c0, c1, c2;
    if (i < NL) {
        const float* p = lidar_coords + ((size_t)b * NL + i) * 3;
        c0 = p[0]; c1 = p[1]; c2 = p[2];
    } else {
        int t = i - NL;
        const float* kuvd = img_kuvd + ((size_t)b * NI + t) * 4;
        int cam = (int)kuvd[0];
        float u = kuvd[1], v = kuvd[2], D = kuvd[3];
        const float* Km = Kc + ((size_t)(b * NVIEW + cam)) * 9;
        const float* Tm = Tc + ((size_t)(b * NVIEW + cam)) * 16;
        const float* Rm = Rp + ((size_t)(b * NVIEW + cam)) * 9;
        const float* tv = tp + ((size_t)(b * NVIEW + cam)) * 3;
        float a0 = u - tv[0], a1 = v - tv[1], a2 = 1.0f - tv[2];
        float Ri[9], Ki[9];
        inv3(Rm, Ri);
        float uv0 = Ri[0]*a0 + Ri[1]*a1 + Ri[2]*a2;
        float uv1 = Ri[3]*a0 + Ri[4]*a1 + Ri[5]*a2;
        inv3(Km, Ki);
        float x0 = (Ki[0]*uv0 + Ki[1]*uv1 + Ki[2]) * D;
        float x1 = (Ki[3]*uv0 + Ki[4]*uv1 + Ki[5]) * D;
        float x2 = (Ki[6]*uv0 + Ki[7]*uv1 + Ki[8]) * D;
        c0 = Tm[0]*x0 + Tm[1]*x1 + Tm[2]*x2  + Tm[3];
        c1 = Tm[4]*x0 + Tm[5]*x1 + Tm[6]*x2  + Tm[7];
        c2 = Tm[8]*x0 + Tm[9]*x1 + Tm[10]*x2 + Tm[11];
    }
    int q0 = (int)floorf(c0), q1 = (int)floorf(c1), q2 = (int)floorf(c2);
    int* qp = q + (size_t)gid * 3;
    qp[0] = q0; qp[1] = q1; qp[2] = q2;

    // ---- hierarchical min/max: wave32 shuffle -> LDS -> 6 atomics per block
    __shared__ int smin[8][3], smax[8][3];
    int wave = threadIdx.x >> 5, lane = threadIdx.x & 31;
    int v0 = q0, v1 = q1, v2 = q2, w0 = q0, w1 = q1, w2 = q2;
    for (int off = 16; off > 0; off >>= 1) {
        v0 = min(v0, __shfl_xor(v0, off));
        v1 = min(v1, __shfl_xor(v1, off));
        v2 = min(v2, __shfl_xor(v2, off));
        w0 = max(w0, __shfl_xor(w0, off));
        w1 = max(w1, __shfl_xor(w1, off));
        w2 = max(w2, __shfl_xor(w2, off));
    }
    if (lane == 0) {
        smin[wave][0] = v0; smin[wave][1] = v1; smin[wave][2] = v2;
        smax[wave][0] = w0; smax[wave][1] = w1; smax[wave][2] = w2;
    }
    __syncthreads();
    if (threadIdx.x < 6) {
        int d = threadIdx.x % 3;
        bool isMax = threadIdx.x >= 3;
        int r = isMax ? INT_MIN : INT_MAX;
        for (int w = 0; w < 8; ++w)
            r = isMax ? max(r, smax[w][d]) : min(r, smin[w][d]);
        if (isMax) atomicMax(&mm[b*6 + 3 + d], r);
        else       atomicMin(&mm[b*6 + d], r);
    }
}

// ============================================================ build keys
__global__ void __launch_bounds__(256)
k_keys(const int* __restrict__ q, const int* __restrict__ mm,
       const unsigned char* __restrict__ lmask, const unsigned char* __restrict__ imask,
       unsigned long long* __restrict__ keys) {
    int gid = blockIdx.x * 256 + threadIdx.x;       // NB*MPAD total
    if (gid >= NB * MPAD) return;
    int b = gid >> LOGM, j = gid & (MPAD - 1);
    if (j >= NPTS) { keys[gid] = ~0ull; return; }   // padding: sorts last
    int mn0 = mm[b*6+0], mn1 = mm[b*6+1], mn2 = mm[b*6+2];
    int mx1 = mm[b*6+4] - mn1, mx2 = mm[b*6+5] - mn2;
    const int* qp = q + ((size_t)b * NPTS + j) * 3;
    int d0 = qp[0] - mn0, d1 = qp[1] - mn1, d2 = qp[2] - mn2;
    int s1 = (d0 & 1) ? (mx1 - d1) : d1;
    int s2 = ((d0 + d1) & 1) ? (mx2 - d2) : d2;
    unsigned long long key = (unsigned long long)d0 * 10000000000ull
                           + (unsigned long long)s1 * 100000ull
                           + (unsigned long long)s2;
    bool msk = (j < NL) ? (lmask[(size_t)b*NL + j] != 0)
                        : (imask[(size_t)b*NI + (j - NL)] != 0);
    if (msk) key = 0xFFFFFFFFFFFFull;               // > any real key, < padding
    keys[gid] = (key << KEYSHIFT) | (unsigned long long)j;  // stable-tie idx in LSBs
}

// ============================================================ fused LDS bitonic
// Runs all passes (j <= TILE/2, intra-tile by construction) for k in
// [kstart..kend] on a TILE-sized segment held in 64KB of LDS. The direction
// bit uses the *global* in-batch index, so this is exact for any k.
__global__ void __launch_bounds__(1024)
k_sort_local(unsigned long long* __restrict__ keys, int kstart, int kend) {
    __shared__ unsigned long long lds[TILE];
    size_t base = (size_t)blockIdx.x * TILE;        // NB*(MPAD/TILE) blocks
    int gbase = (blockIdx.x * TILE) & (MPAD - 1);   // in-batch element base
    for (int t = threadIdx.x; t < TILE; t += 1024) lds[t] = keys[base + t];
    __syncthreads();
    for (int k = kstart; k <= kend; k <<= 1) {
        int j0 = (k >> 1) < (TILE >> 1) ? (k >> 1) : (TILE >> 1);
        for (int j = j0; j >= 1; j >>= 1) {
            for (int i = threadIdx.x; i < TILE; i += 1024) {
                int ixj = i ^ j;
                if (ixj > i) {
                    bool up = (((gbase + i) & k) == 0);
                    unsigned long long a = lds[i], c = lds[ixj];
                    if (up ? (a > c) : (a < c)) { lds[i] = c; lds[ixj] = a; }
                }
            }
            __syncthreads();
        }
    }
    for (int t = threadIdx.x; t < TILE; t += 1024) keys[base + t] = lds[t];
}

// ============================================================ global bitonic pass
__global__ void __launch_bounds__(256)
k_bitonic(unsigned long long* __restrict__ keys, int j, int k) {
    int gid = blockIdx.x * 256 + threadIdx.x;       // NB*MPAD total
    if (gid >= NB * MPAD) return;
    int b = gid >> LOGM, i = gid & (MPAD - 1);
    int ixj = i ^ j;
    if (ixj > i) {
        size_t base = (size_t)b << LOGM;
        unsigned long long a = keys[base + i];
        unsigned long long c = keys[base + ixj];
        bool up = ((i & k) == 0);
        if (up ? (a > c) : (a < c)) { keys[base + i] = c; keys[base + ixj] = a; }
    }
}

// ============================================================ gather + emit
// One wave32 per 1KB token row, streamed global -> LDS -> global through the
// CDNA5 async data-movement path (ASYNCcnt-tracked, no VGPR staging).
// Lane 0 also emits idx (float) and sorted_mask for its row.
__global__ void __launch_bounds__(256)
k_gather(const unsigned long long* __restrict__ keys,
         const unsigned char* __restrict__ lmask, const unsigned char* __restrict__ imask,
         const float* __restrict__ lidar_tokens, const float* __restrict__ img_tokens,
         float* __restrict__ out) {
    __shared__ float stage[8 * CH];                 // 8 waves * 1KB
    int wave = threadIdx.x >> 5;
    int lane = threadIdx.x & 31;
    int row  = blockIdx.x * 8 + wave;               // NB*NPTS rows, grid exact
    int b = row / NPTS, i = row % NPTS;
    unsigned long long p = keys[((size_t)b << LOGM) + i];
    int src = (int)(p & IDXMASK);
    if (lane == 0) {
        out[OUT_IDX + row] = (float)src;
        bool msk = (src < NL) ? (lmask[(size_t)b*NL + src] != 0)
                              : (imask[(size_t)b*NI + (src - NL)] != 0);
        out[OUT_MASK + row] = msk ? 1.0f : 0.0f;
    }
    const float* srow = (src < NL)
        ? lidar_tokens + ((size_t)b * NL + src) * CH
        : img_tokens   + ((size_t)b * NI + (src - NL)) * CH;
    float* drow = out + OUT_TOK + (size_t)row * CH;

    unsigned lds = (unsigned)(size_t)(&stage[wave * CH + lane * 4]); // low 32b = LDS offset
    unsigned long long g  = (unsigned long long)(size_t)(srow + lane * 4);
    unsigned long long dg = (unsigned long long)(size_t)(drow + lane * 4);

    // 2 x b128 per lane covers 1024B per wave (offset applies to LDS and mem)
    asm volatile("global_load_async_to_lds_b128 %0, %1, off"
                 :: "v"(lds), "v"(g) : "memory");
    asm volatile("global_load_async_to_lds_b128 %0, %1, off offset:512"
                 :: "v"(lds), "v"(g) : "memory");
    asm volatile("s_wait_asynccnt 0x0" ::: "memory");
    asm volatile("global_store_async_from_lds_b128 %0, %1, off"
                 :: "v"(dg), "v"(lds) : "memory");
    asm volatile("global_store_async_from_lds_b128 %0, %1, off offset:512"
                 :: "v"(dg), "v"(lds) : "memory");
    asm volatile("s_wait_asynccnt 0x0" ::: "memory");
}

// ============================================================ launch
extern "C" void kernel_launch(void* const* d_in, const int* in_sizes, int n_in,
                              void* d_out, int out_size, void* d_ws, size_t ws_size,
                              hipStream_t stream) {
    const float* lidar_tokens = (const float*)d_in[0];
    const float* lidar_coords = (const float*)d_in[1];
    const float* img_tokens   = (const float*)d_in[2];
    const float* img_kuvd     = (const float*)d_in[3];
    const float* Kc           = (const float*)d_in[4];
    const float* Tc           = (const float*)d_in[5];
    const float* Rp           = (const float*)d_in[6];
    const float* tp           = (const float*)d_in[7];
    const unsigned char* lm   = (const unsigned char*)d_in[8];
    const unsigned char* im   = (const unsigned char*)d_in[9];
    float* out = (float*)d_out;

    char* ws = (char*)d_ws;
    unsigned long long* keys = (unsigned long long*)(ws + WS_KEYS);
    int* q  = (int*)(ws + WS_Q);
    int* mm = (int*)(ws + WS_MM);

    k_init<<<1, 64, 0, stream>>>(mm, out);
    k_quant<<<(NB * NPTS) / 256, 256, 0, stream>>>(lidar_coords, img_kuvd,
                                                   Kc, Tc, Rp, tp, q, mm);
    k_keys<<<(NB * MPAD) / 256, 256, 0, stream>>>(q, mm, lm, im, keys);

    const int nTiles = NB * (MPAD / TILE);          // 16 blocks
    // Phase 1: all k up to TILE fully inside LDS (91 passes, 1 launch)
    k_sort_local<<<nTiles, 1024, 0, stream>>>(keys, 2, TILE);
    // Phase 2: k > TILE -> global passes for j >= TILE, then fused LDS tail
    for (int k = TILE * 2; k <= MPAD; k <<= 1) {
        for (int j = k >> 1; j >= TILE; j >>= 1)
            k_bitonic<<<(NB * MPAD) / 256, 256, 0, stream>>>(keys, j, k);
        k_sort_local<<<nTiles, 1024, 0, stream>>>(keys, k, k);
    }

    k_gather<<<(NB * NPTS) / 8, 256, 0, stream>>>(keys, lm, im,
                                                  lidar_tokens, img_tokens, out);
}